// ParallelUpdate_15642270892348
// MI455X (gfx1250) — compile-verified
//
#include <hip/hip_runtime.h>

typedef __attribute__((ext_vector_type(2))) float v2f;
typedef __attribute__((ext_vector_type(8))) float v8f;

#define D 96
#define HSTRIDE 100  // padded LDS row stride (floats): 100%64=36 -> conflict-free, 400B 16B-aligned

// ---------------- segment_sum: zero + atomic scatter-add (float4 reads) ----------------
__global__ __launch_bounds__(256) void zero_f32(float* __restrict__ p, int n) {
    int i = blockIdx.x * 256 + threadIdx.x;
    if (i < n) p[i] = 0.0f;
}

__global__ __launch_bounds__(256) void scatter_add4(const float4* __restrict__ ea4,
                                                    const int* __restrict__ ei,
                                                    float* __restrict__ agg, int E) {
    long long idx = (long long)blockIdx.x * 256 + threadIdx.x;  // over E * 24 quads
    if (idx >= (long long)E * (D / 4)) return;
    int e = (int)(idx / (D / 4));
    int q = (int)(idx - (long long)e * (D / 4));
    float4 v = ea4[idx];
    int dn = ei[E + e];  // destination node
    float* base = agg + (long long)dn * D + q * 4;
    atomicAdd(base + 0, v.x);
    atomicAdd(base + 1, v.y);
    atomicAdd(base + 2, v.z);
    atomicAdd(base + 3, v.w);
}

// ---------------- weight repack into WMMA B-fragment order ----------------
// W is K x D row-major. Packed: for k-step s, column tile ct, lane l:
//   n = ct*16 + (l&15), kh = l>>4, k = 4s + 2kh
//   Wp[((s*6 + ct)*32 + l)*2 + {0,1}] = W[k][n], W[k+1][n]
// -> each lane's (b0,b1) is one aligned 8-byte load; a wave reads 256B contiguous.
__global__ __launch_bounds__(256) void pack_w(const float* __restrict__ W,
                                              float* __restrict__ Wp, int S) {
    int i = blockIdx.x * 256 + threadIdx.x;  // pair index
    if (i >= S * 6 * 32) return;
    int lane = i & 31;
    int ct   = (i >> 5) % 6;
    int s    = i / 192;
    int n    = ct * 16 + (lane & 15);
    int kh   = lane >> 4;
    int k    = 4 * s + 2 * kh;
    Wp[2 * i]     = W[k * D + n];
    Wp[2 * i + 1] = W[(k + 1) * D + n];
}

// ---------------- edge MLP: h_e = relu([src|dst|ea]@We1+be1)@We2+be2 ----------------
__global__ __launch_bounds__(128) void edge_mlp(
    const float* __restrict__ x, const int* __restrict__ ei,
    const float* __restrict__ ea,
    const float* __restrict__ We1p, const float* __restrict__ be1,
    const float* __restrict__ We2p, const float* __restrict__ be2,
    float* __restrict__ h_e, int E)
{
    __shared__ float lds_h[64 * HSTRIDE];  // 25.6 KB layer-1 activations, 64 rows x 96
    const int tid  = threadIdx.x;
    const int wv   = tid >> 5;        // wave 0..3
    const int lane = tid & 31;
    const int m16  = lane & 15;       // row (A) / col (B) within tile
    const int kh   = lane >> 4;       // K-half selector (0/1)
    const int rowB = wv * 16;
    const int row  = rowB + m16;
    const int eRaw = blockIdx.x * 64 + row;
    const int e    = (eRaw < E) ? eRaw : 0;

    const float* p0 = x  + (long long)ei[e]     * D;  // src feats  (K 0..95)
    const float* p1 = x  + (long long)ei[E + e] * D;  // dst feats  (K 96..191)
    const float* p2 = ea + (long long)e         * D;  // edge feats (K 192..287)

    // ----- layer 1: K = 288 (72 WMMA steps), 6 column tiles -----
    for (int n0 = 0; n0 < D; n0 += 16) {
        const float* bp = We1p + ((n0 >> 4) * 32 + lane) * 2;
        float bias = be1[n0 + m16];
        v8f acc0, acc1;
#pragma unroll
        for (int r = 0; r < 8; ++r) { acc0[r] = bias; acc1[r] = 0.0f; }
#pragma unroll 2
        for (int s = 0; s < 72; s += 2) {
            {
                int k = 4 * s + 2 * kh;
                const float* p = (s < 24) ? p0 : (s < 48) ? p1 : p2;
                int kk = (s < 24) ? k : (s < 48) ? (k - D) : (k - 2 * D);
                v2f a = *(const v2f*)(p + kk);
                v2f b = *(const v2f*)(bp + s * 384);  // 6 tiles * 64 floats per step
                acc0 = __builtin_amdgcn_wmma_f32_16x16x4_f32(
                         false, a, false, b, (short)0, acc0, false, false);
            }
            {
                int s1 = s + 1;
                int k = 4 * s1 + 2 * kh;
                const float* p = (s1 < 24) ? p0 : (s1 < 48) ? p1 : p2;
                int kk = (s1 < 24) ? k : (s1 < 48) ? (k - D) : (k - 2 * D);
                v2f a = *(const v2f*)(p + kk);
                v2f b = *(const v2f*)(bp + s1 * 384);
                acc1 = __builtin_amdgcn_wmma_f32_16x16x4_f32(
                         false, a, false, b, (short)0, acc1, false, false);
            }
        }
#pragma unroll
        for (int r = 0; r < 8; ++r) {  // combine, relu, transpose-stage via LDS
            float v = acc0[r] + acc1[r];
            v = v > 0.0f ? v : 0.0f;
            lds_h[(rowB + r + 8 * kh) * HSTRIDE + n0 + m16] = v;
        }
    }
    __syncthreads();

    // ----- layer 2: K = 96 (24 steps), A from LDS -----
    for (int n0 = 0; n0 < D; n0 += 16) {
        const float* bp = We2p + ((n0 >> 4) * 32 + lane) * 2;
        float bias = be2[n0 + m16];
        v8f acc0, acc1;
#pragma unroll
        for (int r = 0; r < 8; ++r) { acc0[r] = bias; acc1[r] = 0.0f; }
#pragma unroll 2
        for (int s = 0; s < 24; s += 2) {
            {
                int k = 4 * s + 2 * kh;
                v2f a = *(const v2f*)&lds_h[row * HSTRIDE + k];
                v2f b = *(const v2f*)(bp + s * 384);
                acc0 = __builtin_amdgcn_wmma_f32_16x16x4_f32(
                         false, a, false, b, (short)0, acc0, false, false);
            }
            {
                int s1 = s + 1;
                int k = 4 * s1 + 2 * kh;
                v2f a = *(const v2f*)&lds_h[row * HSTRIDE + k];
                v2f b = *(const v2f*)(bp + s1 * 384);
                acc1 = __builtin_amdgcn_wmma_f32_16x16x4_f32(
                         false, a, false, b, (short)0, acc1, false, false);
            }
        }
        const int orow = blockIdx.x * 64 + rowB + 8 * kh;
#pragma unroll
        for (int r = 0; r < 8; ++r) {
            if (orow + r < E)
                h_e[(long long)(orow + r) * D + n0 + m16] = acc0[r] + acc1[r];
        }
    }
}

// ---------------- node MLP: h_v = relu([x|agg]@Wn1+bn1)@Wn2+bn2 ----------------
__global__ __launch_bounds__(128) void node_mlp(
    const float* __restrict__ x, const float* __restrict__ agg,
    const float* __restrict__ Wn1p, const float* __restrict__ bn1,
    const float* __restrict__ Wn2p, const float* __restrict__ bn2,
    float* __restrict__ h_v, int N)
{
    __shared__ float lds_h[64 * HSTRIDE];
    const int tid  = threadIdx.x;
    const int wv   = tid >> 5;
    const int lane = tid & 31;
    const int m16  = lane & 15;
    const int kh   = lane >> 4;
    const int rowB = wv * 16;
    const int row  = rowB + m16;
    const int nRaw = blockIdx.x * 64 + row;
    const int node = (nRaw < N) ? nRaw : 0;   // clamp: loads in-bounds, stores guarded

    const float* p0 = x   + (long long)node * D;  // K 0..95
    const float* p1 = agg + (long long)node * D;  // K 96..191

    for (int n0 = 0; n0 < D; n0 += 16) {
        const float* bp = Wn1p + ((n0 >> 4) * 32 + lane) * 2;
        float bias = bn1[n0 + m16];
        v8f acc0, acc1;
#pragma unroll
        for (int r = 0; r < 8; ++r) { acc0[r] = bias; acc1[r] = 0.0f; }
#pragma unroll 2
        for (int s = 0; s < 48; s += 2) {
            {
                int k = 4 * s + 2 * kh;
                const float* p = (s < 24) ? p0 : p1;
                int kk = (s < 24) ? k : (k - D);
                v2f a = *(const v2f*)(p + kk);
                v2f b = *(const v2f*)(bp + s * 384);
                acc0 = __builtin_amdgcn_wmma_f32_16x16x4_f32(
                         false, a, false, b, (short)0, acc0, false, false);
            }
            {
                int s1 = s + 1;
                int k = 4 * s1 + 2 * kh;
                const float* p = (s1 < 24) ? p0 : p1;
                int kk = (s1 < 24) ? k : (k - D);
                v2f a = *(const v2f*)(p + kk);
                v2f b = *(const v2f*)(bp + s1 * 384);
                acc1 = __builtin_amdgcn_wmma_f32_16x16x4_f32(
                         false, a, false, b, (short)0, acc1, false, false);
            }
        }
#pragma unroll
        for (int r = 0; r < 8; ++r) {
            float v = acc0[r] + acc1[r];
            v = v > 0.0f ? v : 0.0f;
            lds_h[(rowB + r + 8 * kh) * HSTRIDE + n0 + m16] = v;
        }
    }
    __syncthreads();

    for (int n0 = 0; n0 < D; n0 += 16) {
        const float* bp = Wn2p + ((n0 >> 4) * 32 + lane) * 2;
        float bias = bn2[n0 + m16];
        v8f acc0, acc1;
#pragma unroll
        for (int r = 0; r < 8; ++r) { acc0[r] = bias; acc1[r] = 0.0f; }
#pragma unroll 2
        for (int s = 0; s < 24; s += 2) {
            {
                int k = 4 * s + 2 * kh;
                v2f a = *(const v2f*)&lds_h[row * HSTRIDE + k];
                v2f b = *(const v2f*)(bp + s * 384);
                acc0 = __builtin_amdgcn_wmma_f32_16x16x4_f32(
                         false, a, false, b, (short)0, acc0, false, false);
            }
            {
                int s1 = s + 1;
                int k = 4 * s1 + 2 * kh;
                v2f a = *(const v2f*)&lds_h[row * HSTRIDE + k];
                v2f b = *(const v2f*)(bp + s1 * 384);
                acc1 = __builtin_amdgcn_wmma_f32_16x16x4_f32(
                         false, a, false, b, (short)0, acc1, false, false);
            }
        }
        const int orow = blockIdx.x * 64 + rowB + 8 * kh;
#pragma unroll
        for (int r = 0; r < 8; ++r) {
            if (orow + r < N)
                h_v[(long long)(orow + r) * D + n0 + m16] = acc0[r] + acc1[r];
        }
    }
}

extern "C" void kernel_launch(void* const* d_in, const int* in_sizes, int n_in,
                              void* d_out, int out_size, void* d_ws, size_t ws_size,
                              hipStream_t stream) {
    const float* x   = (const float*)d_in[0];
    const int*   ei  = (const int*)d_in[1];
    const float* ea  = (const float*)d_in[2];
    const float* Wn1 = (const float*)d_in[3];
    const float* bn1 = (const float*)d_in[4];
    const float* Wn2 = (const float*)d_in[5];
    const float* bn2 = (const float*)d_in[6];
    const float* We1 = (const float*)d_in[7];
    const float* be1 = (const float*)d_in[8];
    const float* We2 = (const float*)d_in[9];
    const float* be2 = (const float*)d_in[10];

    const int N = in_sizes[0] / D;   // 50000
    const int E = in_sizes[2] / D;   // 800000

    // output tuple layout: h_v [N,D] f32 | edge_index [2,E] | h_e [E,D] f32
    float* h_v    = (float*)d_out;
    char*  ei_out = (char*)d_out + (size_t)N * D * sizeof(float);
    float* h_e    = (float*)(ei_out + (size_t)in_sizes[1] * sizeof(int));

    // workspace layout: agg [N*D] | packed weights
    float* agg  = (float*)d_ws;
    float* Wn1p = agg  + (size_t)N * D;       // S=48: 48*6*64 = 18432 floats
    float* Wn2p = Wn1p + 48 * 6 * 64;         // S=24:  9216 floats
    float* We1p = Wn2p + 24 * 6 * 64;         // S=72: 27648 floats
    float* We2p = We1p + 72 * 6 * 64;         // S=24:  9216 floats

    int aggN = N * D;
    zero_f32<<<(aggN + 255) / 256, 256, 0, stream>>>(agg, aggN);

    long long scat = (long long)E * (D / 4);
    scatter_add4<<<(int)((scat + 255) / 256), 256, 0, stream>>>(
        (const float4*)ea, ei, agg, E);

    pack_w<<<(48 * 192 + 255) / 256, 256, 0, stream>>>(Wn1, Wn1p, 48);
    pack_w<<<(24 * 192 + 255) / 256, 256, 0, stream>>>(Wn2, Wn2p, 24);
    pack_w<<<(72 * 192 + 255) / 256, 256, 0, stream>>>(We1, We1p, 72);
    pack_w<<<(24 * 192 + 255) / 256, 256, 0, stream>>>(We2, We2p, 24);

    node_mlp<<<(N + 63) / 64, 128, 0, stream>>>(x, agg, Wn1p, bn1, Wn2p, bn2, h_v, N);
    edge_mlp<<<(E + 63) / 64, 128, 0, stream>>>(x, ei, ea, We1p, be1, We2p, be2, h_e, E);

    hipMemcpyAsync(ei_out, d_in[1], (size_t)in_sizes[1] * sizeof(int),
                   hipMemcpyDeviceToDevice, stream);
}